// BayesLinear_37950331027759
// MI455X (gfx1250) — compile-verified
//
#include <hip/hip_runtime.h>
#include <math.h>

// ---------------------------------------------------------------------------
// BayesLinear: out = x @ (W_mu + W_rand * softplus(W_rho))^T + (b_mu + b_rand*softplus(b_rho))
// x:[64,4096] fp32, W*:[4096,4096] fp32, out:[64,4096] fp32
// Memory-bound: 201 MB of W tensors vs 2.15 GFLOP -> stream W once, fp32 WMMA,
// native-transcendental softplus, async global->LDS copies for the x tile so
// the register pipeline fits without scratch spills.
// ---------------------------------------------------------------------------

typedef __attribute__((ext_vector_type(2))) float v2f;
typedef __attribute__((ext_vector_type(8))) float v8f;

#define IN_F   4096
#define OUT_F  4096
#define NBATCH 64
#define KC     64          // K-chunk staged in LDS per iteration
#define LSTR   68          // LDS row stride in floats: 16B-aligned, conflict-friendly
#define KSPLIT 4

#if __has_builtin(__builtin_amdgcn_global_load_async_to_lds_b128)
#define HAVE_ASYNC_LDS 1
#else
#define HAVE_ASYNC_LDS 0
#warning "global_load_async_to_lds_b128 builtin unavailable; using sync x path"
#endif

#if HAVE_ASYNC_LDS
// Builtin signature (from hipcc diagnostic): param0 = int4(vector_size 16) addrspace(1)*
typedef int v4i_g __attribute__((vector_size(16)));
typedef __attribute__((address_space(1))) v4i_g* gptr_t;
typedef __attribute__((address_space(3))) v4i_g* lptr_t;
__device__ __forceinline__ void async_copy16(const float* g, float* l) {
    // each lane copies 16 bytes from global address g to LDS address l; tracked by ASYNCcnt
    __builtin_amdgcn_global_load_async_to_lds_b128((gptr_t)g, (lptr_t)l, 0, 0);
}
#endif

__device__ __forceinline__ void wait_async0() {
#if __has_builtin(__builtin_amdgcn_s_wait_asynccnt)
    __builtin_amdgcn_s_wait_asynccnt(0);
#else
    asm volatile("s_wait_asynccnt 0" ::: "memory");
#endif
}

// softplus via native v_exp_f32 / v_log_f32 with a log1p correction term:
//   softplus(r) = max(r,0) + log1p(exp(-|r|))
//   log1p(e) ~= log(t) + (e - (t-1))/t,  t = 1+e   (recovers low-order bits)
__device__ __forceinline__ float softplus_f(float r) {
    float e  = __expf(-fabsf(r));
    float t  = 1.0f + e;
    float lg = __logf(t) + __fdividef(e - (t - 1.0f), t);
    return fmaxf(r, 0.0f) + lg;
}

template <bool WITH_BIAS>
__global__ __launch_bounds__(128)
void bayes_linear_wmma(const float* __restrict__ x,
                       const float* __restrict__ Wmu,
                       const float* __restrict__ Wrho,
                       const float* __restrict__ Wrand,
                       const float* __restrict__ bmu,
                       const float* __restrict__ brho,
                       const float* __restrict__ brand,
                       float* __restrict__ dst_base,
                       int k_per_split)
{
    // double-buffered tiles -> single barrier per chunk
    __shared__ float lds_w[2][16 * LSTR];       // W tile  [16 outs][KC]
    __shared__ float lds_x[2][NBATCH * LSTR];   // x chunk [64 rows][KC]

    const int t     = threadIdx.x;              // 0..127 (4 waves)
    const int lane  = t & 31;
    const int wave  = t >> 5;                   // 0..3 -> batch tile m = 16*wave
    const int obase = blockIdx.x * 16;          // 16 output features per block
    const int split = blockIdx.y;
    const int kbeg  = split * k_per_split;

    float* dst = dst_base + (size_t)split * NBATCH * OUT_F;

    // cooperative-load indexing: 128 threads, 16 threads per row, float4 per thread
    const int lrow = t >> 4;                    // 0..7
    const int lcol = (t & 15) * 4;              // 0,4,...,60

    const size_t wrow0 = (size_t)(obase + lrow)     * IN_F;
    const size_t wrow1 = (size_t)(obase + lrow + 8) * IN_F;

    float4 mu0, mu1, rho0, rho1, rnd0, rnd1;
#if !HAVE_ASYNC_LDS
    float4 xr[8];
#endif

    // ---- prologue: fetch chunk 0 ----
    {
        const int kg = kbeg + lcol;
        mu0  = *(const float4*)(Wmu   + wrow0 + kg);
        mu1  = *(const float4*)(Wmu   + wrow1 + kg);
        rho0 = *(const float4*)(Wrho  + wrow0 + kg);
        rho1 = *(const float4*)(Wrho  + wrow1 + kg);
        rnd0 = *(const float4*)(Wrand + wrow0 + kg);
        rnd1 = *(const float4*)(Wrand + wrow1 + kg);
#pragma unroll
        for (int i = 0; i < 8; ++i) {
#if HAVE_ASYNC_LDS
            async_copy16(x + (size_t)(lrow + 8 * i) * IN_F + kg,
                         &lds_x[0][(lrow + 8 * i) * LSTR + lcol]);
#else
            xr[i] = *(const float4*)(x + (size_t)(lrow + 8 * i) * IN_F + kg);
#endif
        }
    }

    v8f acc = {0.f, 0.f, 0.f, 0.f, 0.f, 0.f, 0.f, 0.f};

    const int nchunk = k_per_split / KC;
    // WMMA fragment addressing (ISA 7.12.2, 32-bit 16x4 A / 4x16 B):
    //   A: lane -> M = lane&15 ; K base = 2*(lane>>4), v2f covers K,K+1
    //   B: lane -> N = lane&15 ; K row  = 2*(lane>>4) + j
    const int mrow  = (wave << 4) + (lane & 15);
    const int nrow  = lane & 15;
    const int khalf = (lane >> 4) << 1;

    for (int c = 0; c < nchunk; ++c) {
        const int buf = c & 1;

        // ---- reparameterize W for this chunk (softplus once per element) ----
        float4 w0, w1;
        w0.x = fmaf(rnd0.x, softplus_f(rho0.x), mu0.x);
        w0.y = fmaf(rnd0.y, softplus_f(rho0.y), mu0.y);
        w0.z = fmaf(rnd0.z, softplus_f(rho0.z), mu0.z);
        w0.w = fmaf(rnd0.w, softplus_f(rho0.w), mu0.w);
        w1.x = fmaf(rnd1.x, softplus_f(rho1.x), mu1.x);
        w1.y = fmaf(rnd1.y, softplus_f(rho1.y), mu1.y);
        w1.z = fmaf(rnd1.z, softplus_f(rho1.z), mu1.z);
        w1.w = fmaf(rnd1.w, softplus_f(rho1.w), mu1.w);

        *(float4*)&lds_w[buf][lrow       * LSTR + lcol] = w0;
        *(float4*)&lds_w[buf][(lrow + 8) * LSTR + lcol] = w1;
#if !HAVE_ASYNC_LDS
#pragma unroll
        for (int i = 0; i < 8; ++i)
            *(float4*)&lds_x[buf][(lrow + 8 * i) * LSTR + lcol] = xr[i];
#endif

        // x tile for this chunk must have landed before the barrier.
        wait_async0();
        // One barrier per chunk: the buffer written above was last *read* two
        // iterations ago, and that read completed before the previous barrier.
        __syncthreads();

        // ---- prefetch next chunk (overlaps WMMA below) ----
        if (c + 1 < nchunk) {
            const int kn  = kbeg + (c + 1) * KC + lcol;
            const int nb  = (c + 1) & 1;
            (void)nb;
#if HAVE_ASYNC_LDS
#pragma unroll
            for (int i = 0; i < 8; ++i)
                async_copy16(x + (size_t)(lrow + 8 * i) * IN_F + kn,
                             &lds_x[nb][(lrow + 8 * i) * LSTR + lcol]);
#else
#pragma unroll
            for (int i = 0; i < 8; ++i)
                xr[i] = *(const float4*)(x + (size_t)(lrow + 8 * i) * IN_F + kn);
#endif
            mu0  = *(const float4*)(Wmu   + wrow0 + kn);
            mu1  = *(const float4*)(Wmu   + wrow1 + kn);
            rho0 = *(const float4*)(Wrho  + wrow0 + kn);
            rho1 = *(const float4*)(Wrho  + wrow1 + kn);
            rnd0 = *(const float4*)(Wrand + wrow0 + kn);
            rnd1 = *(const float4*)(Wrand + wrow1 + kn);
        }

        // ---- fp32 matrix core: 16 x v_wmma_f32_16x16x4_f32 per chunk ----
#pragma unroll
        for (int kk = 0; kk < KC; kk += 4) {
            v2f a = *(const v2f*)&lds_x[buf][mrow * LSTR + kk + khalf];
            v2f b = *(const v2f*)&lds_w[buf][nrow * LSTR + kk + khalf];
            acc = __builtin_amdgcn_wmma_f32_16x16x4_f32(
                false, a, false, b, (short)0, acc, false, false);
        }
    }

    // ---- epilogue: C/D layout -> global (VGPR v: M = v + 8*(lane>=16)) ----
    const int o = obase + (lane & 15);
    float bias = 0.0f;
    if (WITH_BIAS)
        bias = fmaf(brand[o], softplus_f(brho[o]), bmu[o]);
    const int mtop = (wave << 4) + ((lane >> 4) << 3);
#pragma unroll
    for (int v = 0; v < 8; ++v)
        dst[(size_t)(mtop + v) * OUT_F + o] = acc[v] + bias;
}

// Deterministic k-split reduction + bias (fixed summation order; no fp atomics).
__global__ __launch_bounds__(256)
void reduce_bias_kernel(const float* __restrict__ part,
                        const float* __restrict__ bmu,
                        const float* __restrict__ brho,
                        const float* __restrict__ brand,
                        float* __restrict__ out, int nsplit)
{
    const int i = blockIdx.x * 256 + threadIdx.x;
    if (i >= NBATCH * OUT_F) return;
    float s = 0.0f;
    for (int j = 0; j < nsplit; ++j)
        s += part[(size_t)j * NBATCH * OUT_F + i];
    const int o = i & (OUT_F - 1);
    out[i] = s + fmaf(brand[o], softplus_f(brho[o]), bmu[o]);
}

extern "C" void kernel_launch(void* const* d_in, const int* in_sizes, int n_in,
                              void* d_out, int out_size, void* d_ws, size_t ws_size,
                              hipStream_t stream)
{
    (void)in_sizes; (void)n_in; (void)out_size;
    // setup_inputs order: x, W_mu, W_rho, b_mu, b_rho, W_rand, b_rand
    const float* x     = (const float*)d_in[0];
    const float* Wmu   = (const float*)d_in[1];
    const float* Wrho  = (const float*)d_in[2];
    const float* bmu   = (const float*)d_in[3];
    const float* brho  = (const float*)d_in[4];
    const float* Wrand = (const float*)d_in[5];
    const float* brand = (const float*)d_in[6];
    float* out = (float*)d_out;

    const dim3 block(128);
    const size_t need = (size_t)KSPLIT * NBATCH * OUT_F * sizeof(float);

    if (ws_size >= need) {
        // 4-way K split for concurrency: 1024 workgroups streaming W.
        float* part = (float*)d_ws;
        dim3 grid(OUT_F / 16, KSPLIT);
        bayes_linear_wmma<false><<<grid, block, 0, stream>>>(
            x, Wmu, Wrho, Wrand, bmu, brho, brand, part, IN_F / KSPLIT);
        const int n = NBATCH * OUT_F;
        reduce_bias_kernel<<<(n + 255) / 256, 256, 0, stream>>>(
            part, bmu, brho, brand, out, KSPLIT);
    } else {
        // Fallback: single pass, bias fused in epilogue.
        dim3 grid(OUT_F / 16, 1);
        bayes_linear_wmma<true><<<grid, block, 0, stream>>>(
            x, Wmu, Wrho, Wrand, bmu, brho, brand, out, IN_F);
    }
}